// EnergyContrastiveLoss_40269613367603
// MI455X (gfx1250) — compile-verified
//
#include <hip/hip_runtime.h>

typedef __attribute__((ext_vector_type(2))) float v2f;
typedef __attribute__((ext_vector_type(8))) float v8f;
typedef __attribute__((ext_vector_type(2))) int   v2i;

#define B_N    8192
#define P_N    16
#define MARGIN 1.0f

// Pair stage tiling: 256 threads/block, 4 i-rows per thread, 256 j per block.
#define IPT    4
#define IBLKR  (256 * IPT)          // 1024 i-rows per block
#define JBLK   256                  // j-columns per block (LDS resident)
#define IGRID  (B_N / IBLKR)        // 8
#define JGRID  (B_N / JBLK)         // 32
#define NPART  (IGRID * JGRID)      // 256 partials

// Async global->LDS path (ASYNCcnt-tracked), guarded so the file always builds.
#if defined(__gfx1250__) && \
    __has_builtin(__builtin_amdgcn_global_load_async_to_lds_b64) && \
    __has_builtin(__builtin_amdgcn_s_wait_asynccnt)
#define ECL_HAS_ASYNC 1
#else
#define ECL_HAS_ASYNC 0
#endif

// ---------------------------------------------------------------------------
// Stage 1: d2e[i] = { ||pv[i,:] - t||^2 , energies[i] }  (packed float2)
// One wave per 16-row tile; diag(D*D^T) via 4 chained V_WMMA_F32_16X16X4_F32,
// exploiting that the f32 A(16x4) and B(4x16)=A^T register images coincide.
// ---------------------------------------------------------------------------
__global__ __launch_bounds__(256) void ecl_dist2_wmma(
    const float* __restrict__ pv,
    const float* __restrict__ tgt,
    const float* __restrict__ energies,
    float2* __restrict__ d2e)
{
    const int lane = threadIdx.x & 31;
    const int wave = threadIdx.x >> 5;
    const int tile = blockIdx.x * 8 + wave;   // 16 rows per tile
    const int row0 = tile * 16;
    const int m    = lane & 15;               // row index within tile
    const int h    = lane >> 4;               // which K-pair this lane holds

    const float* rowp = pv + (size_t)(row0 + m) * P_N;

    v8f acc = {};
#pragma unroll
    for (int c = 0; c < 4; ++c) {
        const int k0 = 4 * c + 2 * h;
        v2f d;
        d.x = rowp[k0]     - tgt[k0];
        d.y = rowp[k0 + 1] - tgt[k0 + 1];
        // D * D^T accumulation: same registers serve as A and B.
        acc = __builtin_amdgcn_wmma_f32_16x16x4_f32(
                  false, d, false, d, (short)0, acc, false, false);
    }

    // Gram diagonal lives at: lanes 0..7 -> diag(lane) = acc[lane],
    //                         lanes 24..31 -> diag(lane-16) = acc[lane-24].
    int sel = -1, di = 0;
    if (lane < 8)        { sel = lane;      di = lane;      }
    else if (lane >= 24) { sel = lane - 24; di = lane - 16; }

    if (sel >= 0) {
        float dval = acc[0];
        dval = (sel == 1) ? acc[1] : dval;
        dval = (sel == 2) ? acc[2] : dval;
        dval = (sel == 3) ? acc[3] : dval;
        dval = (sel == 4) ? acc[4] : dval;
        dval = (sel == 5) ? acc[5] : dval;
        dval = (sel == 6) ? acc[6] : dval;
        dval = (sel == 7) ? acc[7] : dval;
        d2e[row0 + di] = make_float2(dval, energies[row0 + di]);
    }
}

// ---------------------------------------------------------------------------
// Stage 2: tiled O(B^2) masked pair loss, 4 i-rows register-blocked per
// thread so each LDS float2 broadcast feeds 4 pair updates (~16 VALU / ds op).
// j-chunk staged into LDS via async global->LDS b64 when available.
// ---------------------------------------------------------------------------
__global__ __launch_bounds__(256) void ecl_pair(
    const float2* __restrict__ d2e,
    float* __restrict__ psum,
    unsigned* __restrict__ pcount)
{
    __shared__ float2   sj[JBLK];
    __shared__ float    rs[256];
    __shared__ unsigned rc[256];

    const int tid = threadIdx.x;
    const int ib  = blockIdx.x;
    const int jb  = blockIdx.y;
    const int j0  = jb * JBLK;

#if ECL_HAS_ASYNC
    // One ASYNCcnt-tracked b64 per lane: global d2e chunk -> LDS, no VGPR hop.
    __builtin_amdgcn_global_load_async_to_lds_b64(
        (v2i*)(d2e + j0 + tid),
        (__attribute__((address_space(3))) v2i*)(&sj[tid]),
        0, 0);
    __builtin_amdgcn_s_wait_asynccnt(0);
    __syncthreads();
#else
    sj[tid] = d2e[j0 + tid];
    __syncthreads();
#endif

    float    d2i[IPT], eim[IPT], lsum[IPT];
    unsigned lcnt[IPT];
#pragma unroll
    for (int r = 0; r < IPT; ++r) {
        const float2 mine = d2e[ib * IBLKR + r * 256 + tid];
        d2i[r]  = mine.x;
        eim[r]  = mine.y + MARGIN;
        lsum[r] = 0.0f;
        lcnt[r] = 0u;
    }

#pragma unroll 2
    for (int t = 0; t < JBLK; ++t) {
        const float2 je = sj[t];               // LDS broadcast read
#pragma unroll
        for (int r = 0; r < IPT; ++r) {
            const bool  msk = d2i[r] < je.x;   // == (||.||_i < ||.||_j)
            const float pl  = fmaxf(eim[r] - je.y, 0.0f);
            lsum[r] += msk ? pl : 0.0f;
            lcnt[r] += msk ? 1u : 0u;
        }
    }

    // Fixed-order per-thread combine, then fixed-order block tree reduce.
    rs[tid] = (lsum[0] + lsum[1]) + (lsum[2] + lsum[3]);
    rc[tid] = (lcnt[0] + lcnt[1]) + (lcnt[2] + lcnt[3]);
    __syncthreads();
#pragma unroll
    for (int s = 128; s > 0; s >>= 1) {
        if (tid < s) { rs[tid] += rs[tid + s]; rc[tid] += rc[tid + s]; }
        __syncthreads();
    }
    if (tid == 0) {
        const int bid = ib * JGRID + jb;
        psum[bid]   = rs[0];
        pcount[bid] = rc[0];
    }
}

// ---------------------------------------------------------------------------
// Stage 3: fixed-order reduction of the 256 block partials -> loss.
// ---------------------------------------------------------------------------
__global__ __launch_bounds__(256) void ecl_finalize(
    const float* __restrict__ psum,
    const unsigned* __restrict__ pcount,
    float* __restrict__ out)
{
    __shared__ float    rs[256];
    __shared__ unsigned rc[256];
    const int tid = threadIdx.x;
    rs[tid] = psum[tid];
    rc[tid] = pcount[tid];
    __syncthreads();
#pragma unroll
    for (int s = 128; s > 0; s >>= 1) {
        if (tid < s) { rs[tid] += rs[tid + s]; rc[tid] += rc[tid + s]; }
        __syncthreads();
    }
    if (tid == 0) {
        const float cnt = (float)rc[0];
        out[0] = rs[0] / fmaxf(cnt, 1.0f);
    }
}

// ---------------------------------------------------------------------------
extern "C" void kernel_launch(void* const* d_in, const int* in_sizes, int n_in,
                              void* d_out, int out_size, void* d_ws, size_t ws_size,
                              hipStream_t stream) {
    const float* energies = (const float*)d_in[0];   // (8192, 1)
    const float* pv       = (const float*)d_in[1];   // (8192, 16)
    const float* tgt      = (const float*)d_in[2];   // (16,)
    float* out = (float*)d_out;

    // Workspace layout: float2 d2e[8192] | float psum[256] | u32 pcount[256]
    float2*   d2e    = (float2*)d_ws;
    float*    psum   = (float*)(d2e + B_N);
    unsigned* pcount = (unsigned*)(psum + NPART);

    // Stage 1: 64 blocks * 8 waves * 16 rows = 8192 rows
    ecl_dist2_wmma<<<64, 256, 0, stream>>>(pv, tgt, energies, d2e);

    // Stage 2: 8 x 32 = 256 blocks
    ecl_pair<<<dim3(IGRID, JGRID), 256, 0, stream>>>(d2e, psum, pcount);

    // Stage 3
    ecl_finalize<<<1, 256, 0, stream>>>(psum, pcount, out);
}